// GNNHeuristic_12008728560246
// MI455X (gfx1250) — compile-verified
//
#include <hip/hip_runtime.h>

typedef __attribute__((ext_vector_type(2))) float v2f;
typedef __attribute__((ext_vector_type(8))) float v8f;

// ---------------------------------------------------------------------------
// Utility: zero a float buffer
// ---------------------------------------------------------------------------
__global__ void k_zero(float* __restrict__ p, long long n) {
  long long i = (long long)blockIdx.x * blockDim.x + threadIdx.x;
  if (i < n) p[i] = 0.0f;
}

// ---------------------------------------------------------------------------
// h0 = node_feats @ Wn.T + bn   (N,3)x(3,64)
// ---------------------------------------------------------------------------
__global__ void k_node_fc(const float* __restrict__ nf, const float* __restrict__ Wn,
                          const float* __restrict__ bn, float* __restrict__ h, int N) {
  int t = blockIdx.x * blockDim.x + threadIdx.x;
  int n = t >> 6, j = t & 63;
  if (n >= N) return;
  float f0 = nf[n * 3 + 0], f1 = nf[n * 3 + 1], f2 = nf[n * 3 + 2];
  h[t] = bn[j] + Wn[j * 3 + 0] * f0 + Wn[j * 3 + 1] * f1 + Wn[j * 3 + 2] * f2;
}

// ---------------------------------------------------------------------------
// degree counts (float, exact for counts < 2^24)
// ---------------------------------------------------------------------------
__global__ void k_degree(const int* __restrict__ src, float* __restrict__ cnt, int E) {
  int e = blockIdx.x * blockDim.x + threadIdx.x;
  if (e < E) atomicAdd(&cnt[src[e]], 1.0f);
}

// ---------------------------------------------------------------------------
// accum[src[e]] += h[dst[e]]   (segment sum, L2-resident atomics)
// one thread per (edge, 16-column slice)
// ---------------------------------------------------------------------------
__global__ void k_scatter(const int* __restrict__ src, const int* __restrict__ dst,
                          const float* __restrict__ h, float* __restrict__ accum, int E) {
  int t = blockIdx.x * blockDim.x + threadIdx.x;
  int e = t >> 2;
  if (e >= E) return;
  int q = (t & 3) * 16;
  const float* hr = h + (long long)dst[e] * 64 + q;
  float* ar = accum + (long long)src[e] * 64 + q;
#pragma unroll
  for (int i = 0; i < 16; ++i) atomicAdd(&ar[i], hr[i]);
}

// ---------------------------------------------------------------------------
// h_out = relu([h | accum/deg] (Nx128) @ [Wself|Wneigh].T + bself + bneigh)
// WMMA f32 16x16x4, one 16-node tile per wave (8 waves / block)
// B (K=128 -> 32 chunks, 4 col-tiles) pre-swizzled into LDS.
// ---------------------------------------------------------------------------
__global__ void __launch_bounds__(256)
k_layer_gemm(const float* __restrict__ hin, const float* __restrict__ accum,
             const float* __restrict__ cnt,
             const float* __restrict__ Wself, const float* __restrict__ bself,
             const float* __restrict__ Wneigh, const float* __restrict__ bneigh,
             float* __restrict__ hout, int N) {
  __shared__ float Bl[32 * 4 * 32 * 2];  // 32 KB: [kk][ct][lane][v]
  int tid = threadIdx.x;
  for (int idx = tid; idx < 8192; idx += 256) {
    int v = idx & 1, lane = (idx >> 1) & 31, ct = (idx >> 6) & 3, kk = idx >> 8;
    int n = ct * 16 + (lane & 15);
    int k = kk * 4 + 2 * (lane >> 4) + v;
    Bl[idx] = (k < 64) ? Wself[n * 64 + k] : Wneigh[n * 64 + (k - 64)];
  }
  __syncthreads();

  int w = tid >> 5, lane = tid & 31;
  int base = (blockIdx.x * 8 + w) * 16;
  int half = lane >> 4;
  int mrow = base + (lane & 15);
  long long mc = (mrow < N) ? mrow : 0;

  float c = cnt[mc];
  float inv = (c > 0.f) ? 1.f / c : 1.f;

  // A fragments: lane L holds row L%16, K-pair 2*(L/16)+{0,1} of each 4-chunk
  v2f A[32];
#pragma unroll
  for (int kk = 0; kk < 16; ++kk) {
    int k0 = kk * 4 + 2 * half;
    v2f a = {hin[mc * 64 + k0], hin[mc * 64 + k0 + 1]};
    A[kk] = a;
  }
#pragma unroll
  for (int kk = 0; kk < 16; ++kk) {
    int k0 = kk * 4 + 2 * half;
    v2f a = {accum[mc * 64 + k0] * inv, accum[mc * 64 + k0 + 1] * inv};
    A[16 + kk] = a;
  }

  v8f acc0 = {}, acc1 = {}, acc2 = {}, acc3 = {};
#pragma unroll
  for (int kk = 0; kk < 32; ++kk) {
    int bb = kk * 256 + lane * 2;
    acc0 = __builtin_amdgcn_wmma_f32_16x16x4_f32(false, A[kk], false, *(const v2f*)&Bl[bb + 0],   (short)0, acc0, false, false);
    acc1 = __builtin_amdgcn_wmma_f32_16x16x4_f32(false, A[kk], false, *(const v2f*)&Bl[bb + 64],  (short)0, acc1, false, false);
    acc2 = __builtin_amdgcn_wmma_f32_16x16x4_f32(false, A[kk], false, *(const v2f*)&Bl[bb + 128], (short)0, acc2, false, false);
    acc3 = __builtin_amdgcn_wmma_f32_16x16x4_f32(false, A[kk], false, *(const v2f*)&Bl[bb + 192], (short)0, acc3, false, false);
  }

#pragma unroll
  for (int ct = 0; ct < 4; ++ct) {
    v8f av = (ct == 0) ? acc0 : (ct == 1) ? acc1 : (ct == 2) ? acc2 : acc3;
    int n = ct * 16 + (lane & 15);
    float bs = bself[n] + bneigh[n];
#pragma unroll
    for (int r = 0; r < 8; ++r) {
      int mg = base + r + 8 * half;  // C layout: VGPR r holds M = r + 8*(lane/16)
      if (mg < N) {
        float val = av[r] + bs;
        hout[(long long)mg * 64 + n] = val > 0.f ? val : 0.f;
      }
    }
  }
}

// ---------------------------------------------------------------------------
// Fold the edge path:  Afold(64x3) = Wa[:, :64] @ We
//                      cfold(64)   = Wa[:, :64]@be + ba + Wa[:, 64:]@vnf_mean
// ---------------------------------------------------------------------------
__global__ void k_prep(const float* __restrict__ Wa, const float* __restrict__ ba,
                       const float* __restrict__ We, const float* __restrict__ be,
                       const float* __restrict__ env, int NV,
                       float* __restrict__ afold, float* __restrict__ cfold) {
  int j = threadIdx.x;
  if (j >= 64) return;
  float a0 = 0.f, a1 = 0.f, a2 = 0.f, c = ba[j];
  for (int k = 0; k < 64; ++k) {
    float wv = Wa[j * 128 + k];
    a0 += wv * We[k * 3 + 0];
    a1 += wv * We[k * 3 + 1];
    a2 += wv * We[k * 3 + 2];
    c += wv * be[k];
  }
  float invn = 1.0f / (float)NV;
  for (int d = 0; d < 64; ++d) {
    float vm = 0.f;
    for (int v = 0; v < NV; ++v) vm += env[v * 64 + d];
    c += Wa[j * 128 + 64 + d] * (vm * invn);
  }
  afold[j * 3 + 0] = a0;
  afold[j * 3 + 1] = a1;
  afold[j * 3 + 2] = a2;
  cfold[j] = c;
}

// ---------------------------------------------------------------------------
// scores[e] = W2 . relu(W1 @ relu(Afold@f + cfold) + b1) + b2
// one 16-edge tile per wave; 64 wmma f32_16x16x4 per tile (K=64, 4 col tiles)
// ---------------------------------------------------------------------------
__global__ void __launch_bounds__(256)
k_edge_score(const float* __restrict__ ef, const float* __restrict__ afold,
             const float* __restrict__ cfold, const float* __restrict__ W1,
             const float* __restrict__ b1, const float* __restrict__ W2,
             const float* __restrict__ b2, float* __restrict__ scores, int E) {
  __shared__ float Bs[16 * 4 * 32 * 2];  // 16 KB: W1.T swizzled [kk][ct][lane][v]
  __shared__ float Af[192];
  __shared__ float Cf[64];
  int tid = threadIdx.x;
  for (int idx = tid; idx < 4096; idx += 256) {
    int v = idx & 1, lane = (idx >> 1) & 31, ct = (idx >> 6) & 3, kk = idx >> 8;
    int n = ct * 16 + (lane & 15);
    int k = kk * 4 + 2 * (lane >> 4) + v;
    Bs[idx] = W1[n * 64 + k];
  }
  if (tid < 192) Af[tid] = afold[tid];
  if (tid < 64) Cf[tid] = cfold[tid];
  __syncthreads();

  int w = tid >> 5, lane = tid & 31;
  int base = (blockIdx.x * 8 + w) * 16;
  int half = lane >> 4;
  int row = base + (lane & 15);
  long long rc = (row < E) ? row : 0;
  float f0 = ef[rc * 3 + 0], f1 = ef[rc * 3 + 1], f2 = ef[rc * 3 + 2];

  // Build activation tile directly in WMMA A layout (fused 3->64 affine + relu)
  v2f A[16];
#pragma unroll
  for (int kk = 0; kk < 16; ++kk) {
    int k0 = kk * 4 + 2 * half;
    float x0 = Cf[k0] + Af[k0 * 3] * f0 + Af[k0 * 3 + 1] * f1 + Af[k0 * 3 + 2] * f2;
    float x1 = Cf[k0 + 1] + Af[(k0 + 1) * 3] * f0 + Af[(k0 + 1) * 3 + 1] * f1 +
               Af[(k0 + 1) * 3 + 2] * f2;
    v2f a = {x0 > 0.f ? x0 : 0.f, x1 > 0.f ? x1 : 0.f};
    A[kk] = a;
  }

  v8f acc0 = {}, acc1 = {}, acc2 = {}, acc3 = {};
#pragma unroll
  for (int kk = 0; kk < 16; ++kk) {
    int bb = kk * 256 + lane * 2;
    acc0 = __builtin_amdgcn_wmma_f32_16x16x4_f32(false, A[kk], false, *(const v2f*)&Bs[bb + 0],   (short)0, acc0, false, false);
    acc1 = __builtin_amdgcn_wmma_f32_16x16x4_f32(false, A[kk], false, *(const v2f*)&Bs[bb + 64],  (short)0, acc1, false, false);
    acc2 = __builtin_amdgcn_wmma_f32_16x16x4_f32(false, A[kk], false, *(const v2f*)&Bs[bb + 128], (short)0, acc2, false, false);
    acc3 = __builtin_amdgcn_wmma_f32_16x16x4_f32(false, A[kk], false, *(const v2f*)&Bs[bb + 192], (short)0, acc3, false, false);
  }

  // fused bias + relu + W2 dot; per-lane partial over its 4 columns
  float p[8] = {0, 0, 0, 0, 0, 0, 0, 0};
#pragma unroll
  for (int ct = 0; ct < 4; ++ct) {
    v8f av = (ct == 0) ? acc0 : (ct == 1) ? acc1 : (ct == 2) ? acc2 : acc3;
    int n = ct * 16 + (lane & 15);
    float bn_ = b1[n], wn = W2[n];
#pragma unroll
    for (int r = 0; r < 8; ++r) {
      float s = av[r] + bn_;
      s = s > 0.f ? s : 0.f;
      p[r] += wn * s;
    }
  }
  // reduce across each 16-lane half (rows 0-7 in lanes 0-15, rows 8-15 in 16-31)
#pragma unroll
  for (int r = 0; r < 8; ++r) {
    p[r] += __shfl_xor(p[r], 1);
    p[r] += __shfl_xor(p[r], 2);
    p[r] += __shfl_xor(p[r], 4);
    p[r] += __shfl_xor(p[r], 8);
  }
  if ((lane & 15) == 0) {
    float bb2 = b2[0];
    int rb = base + 8 * half;
#pragma unroll
    for (int r = 0; r < 8; ++r)
      if (rb + r < E) scores[rb + r] = p[r] + bb2;
  }
}

// ---------------------------------------------------------------------------
extern "C" void kernel_launch(void* const* d_in, const int* in_sizes, int n_in,
                              void* d_out, int out_size, void* d_ws, size_t ws_size,
                              hipStream_t stream) {
  if (n_in < 18) return;
  const float* node_feats = (const float*)d_in[0];
  const float* edge_feats = (const float*)d_in[1];
  const float* env        = (const float*)d_in[2];
  const int*   eidx       = (const int*)d_in[3];
  const float* Wn  = (const float*)d_in[4];
  const float* bn  = (const float*)d_in[5];
  const float* We  = (const float*)d_in[6];
  const float* be  = (const float*)d_in[7];
  const float* Wself  = (const float*)d_in[8];
  const float* bself  = (const float*)d_in[9];
  const float* Wneigh = (const float*)d_in[10];
  const float* bneigh = (const float*)d_in[11];
  const float* Wa  = (const float*)d_in[12];
  const float* ba  = (const float*)d_in[13];
  const float* W1  = (const float*)d_in[14];
  const float* b1  = (const float*)d_in[15];
  const float* W2  = (const float*)d_in[16];
  const float* b2  = (const float*)d_in[17];

  int N  = in_sizes[0] / 3;
  int E  = in_sizes[1] / 3;
  int NV = in_sizes[2] / 64;
  int L  = in_sizes[8] / (64 * 64);

  const int* src = eidx;
  const int* dst = eidx + E;

  float* scores = (float*)d_out;
  float* h_out  = (float*)d_out + E;  // N*64 (final h; also reused as h0)

  float* accum = (float*)d_ws;                       // N*64
  float* cnt   = accum + (long long)N * 64;          // N
  long long off = (((long long)N * 65) + 63) & ~63LL;
  float* h_b   = (float*)d_ws + off;                 // N*64 ping buffer
  float* afold = h_b + (long long)N * 64;            // 192
  float* cfold = afold + 192;                        // 64

  // zero accum + cnt (contiguous)
  long long zn = (long long)N * 65;
  k_zero<<<(int)((zn + 255) / 256), 256, 0, stream>>>(accum, zn);
  // node_fc -> h0 (in d_out h region)
  k_node_fc<<<(N * 64 + 255) / 256, 256, 0, stream>>>(node_feats, Wn, bn, h_out, N);
  // degrees
  k_degree<<<(E + 255) / 256, 256, 0, stream>>>(src, cnt, E);

  int tiles = (N + 15) / 16;
  float* hin = h_out;
  for (int l = 0; l < L; ++l) {
    if (l > 0) {
      long long an = (long long)N * 64;
      k_zero<<<(int)((an + 255) / 256), 256, 0, stream>>>(accum, an);
    }
    k_scatter<<<(int)(((long long)E * 4 + 255) / 256), 256, 0, stream>>>(src, dst, hin, accum, E);
    // in-place per-tile update is safe (each wave reads only its own 16 rows
    // before writing them), so the last layer may write h_out even if hin==h_out
    float* hout = (l == L - 1) ? h_out : ((hin == h_out) ? h_b : h_out);
    k_layer_gemm<<<(tiles + 7) / 8, 256, 0, stream>>>(
        hin, accum, cnt,
        Wself + (long long)l * 64 * 64, bself + (long long)l * 64,
        Wneigh + (long long)l * 64 * 64, bneigh + (long long)l * 64, hout, N);
    hin = hout;
  }

  // folded edge-path weights, then fused edge scorer
  k_prep<<<1, 64, 0, stream>>>(Wa, ba, We, be, env, NV, afold, cfold);
  int etiles = (E + 15) / 16;
  k_edge_score<<<(etiles + 7) / 8, 256, 0, stream>>>(edge_feats, afold, cfold, W1, b1, W2,
                                                     b2, scores, E);
}